// TemporalDecoder_48644799594822
// MI455X (gfx1250) — compile-verified
//
#include <hip/hip_runtime.h>
#include <hip/hip_bf16.h>
#include <math.h>

typedef float v2f __attribute__((ext_vector_type(2)));
typedef float v8f __attribute__((ext_vector_type(8)));

__device__ __forceinline__ float gelu_erf(float x) {
    // torch default GELU (erf form)
    return 0.5f * x * (1.0f + erff(x * 0.70710678118654752440f));
}

// =====================================================================
// f32 WMMA GEMM:  D = act(A @ B + bias)
// A: MxK row-major (lda=K), B: KxN row-major (ldb), D: row-major (ldd)
// Block = 256 threads (8 waves), tile 64x128, BK=32, double-buffered LDS.
// Wave (wm 0..1, wn 0..3) owns a 32x32 slab = four 16x16 f32 accumulators
// -> 4 WMMA per 4 fragment loads, every fragment reused twice.
// f32 WMMA lane layout (ISA 7.12.2):
//   A 16x4 : lane l holds A[l&15][2*(l>>4) + {0,1}]   (k-pair contiguous)
//   B 4x16 : lane l holds B[2*(l>>4) + {0,1}][l&15]
//   C 16x16: VGPR r, lanes 0-15 -> M=r, lanes 16-31 -> M=r+8, N=l&15
// B is staged in a fragment-ready paired layout:
//   Bs[kp][n*2 + (k&1)] = B[k][n]  ->  one ds_load_b64 per B fragment.
// Row strides: As 36 (conflict-free), Bs 288 (== 32 mod 64: half-waves
// reading kp / kp+1 land on disjoint 32-bank halves).
// =====================================================================
template<bool GELU>
__global__ __launch_bounds__(256)
void gemm_f32_wmma(const float* __restrict__ A, const float* __restrict__ Bw,
                   const float* __restrict__ bias, float* __restrict__ D,
                   int K, int ldb, int ldd, int Nstore)
{
    __shared__ float As[2][64][36];
    __shared__ float Bs[2][16][288];

    const int tid  = threadIdx.x;
    const int lane = tid & 31;
    const int w    = tid >> 5;       // 0..7
    const int wm   = w & 1;          // row sub-tile (32 rows)
    const int wn   = w >> 1;         // col sub-tile (32 cols)
    const int row0 = blockIdx.x * 64;
    const int col0 = blockIdx.y * 128;

    const int m16  = lane & 15;
    const int half = lane >> 4;      // 0 or 1
    const int khi  = half << 1;      // 0 or 2

    // A staging coords (float4 granularity): 64x32 tile
    const int ar = tid >> 3;             // 0..31
    const int ac = (tid & 7) << 2;       // 0..28
    // B staging coords: 32x128 tile, 4 float4 per thread
    const int brow = tid >> 3;           // 0..31
    const int bcg  = (tid & 7) << 2;     // 0..28
    const int bkp  = brow >> 1;
    const int bpar = brow & 1;

    v8f c00 = {}, c01 = {}, c10 = {}, c11 = {};

    auto stage = [&](int k0, int buf) {
        float4 va0 = *(const float4*)(A + (size_t)(row0 + ar)      * K + k0 + ac);
        float4 va1 = *(const float4*)(A + (size_t)(row0 + ar + 32) * K + k0 + ac);
        *(float4*)(&As[buf][ar][ac])      = va0;
        *(float4*)(&As[buf][ar + 32][ac]) = va1;
        #pragma unroll
        for (int j = 0; j < 4; ++j) {
            const int col = bcg + j * 32;
            float4 vb = *(const float4*)(Bw + (size_t)(k0 + brow) * ldb + col0 + col);
            Bs[buf][bkp][((col + 0) << 1) | bpar] = vb.x;
            Bs[buf][bkp][((col + 1) << 1) | bpar] = vb.y;
            Bs[buf][bkp][((col + 2) << 1) | bpar] = vb.z;
            Bs[buf][bkp][((col + 3) << 1) | bpar] = vb.w;
        }
    };

    stage(0, 0);
    __syncthreads();

    const int nchunks = K >> 5;
    for (int ch = 0; ch < nchunks; ++ch) {
        const int buf = ch & 1;
        if (ch + 1 < nchunks) stage((ch + 1) << 5, buf ^ 1);

        const float* Ar0 = &As[buf][wm * 32 + m16][0];
        const float* Ar1 = &As[buf][wm * 32 + 16 + m16][0];
        #pragma unroll
        for (int kk = 0; kk < 32; kk += 4) {
            const int kb = kk + khi;
            v2f a0 = *(const v2f*)(Ar0 + kb);
            v2f a1 = *(const v2f*)(Ar1 + kb);
            const float* bp = &Bs[buf][(kk >> 1) + half][0];
            v2f b0 = *(const v2f*)(bp + ((wn * 32 + m16) << 1));
            v2f b1 = *(const v2f*)(bp + ((wn * 32 + 16 + m16) << 1));
            c00 = __builtin_amdgcn_wmma_f32_16x16x4_f32(false, a0, false, b0,
                                                        (short)0, c00, false, false);
            c01 = __builtin_amdgcn_wmma_f32_16x16x4_f32(false, a0, false, b1,
                                                        (short)0, c01, false, false);
            c10 = __builtin_amdgcn_wmma_f32_16x16x4_f32(false, a1, false, b0,
                                                        (short)0, c10, false, false);
            c11 = __builtin_amdgcn_wmma_f32_16x16x4_f32(false, a1, false, b1,
                                                        (short)0, c11, false, false);
        }
        __syncthreads();
    }

    // epilogue: bias (+GELU), predicated store for the N-padded case
    const int rb0 = row0 + wm * 32 + (half << 3);
    const int nc0 = col0 + wn * 32 + m16;
    #pragma unroll
    for (int r = 0; r < 8; ++r) {
        const int rlo = rb0 + r;
        const int rhi = rb0 + 16 + r;
        int n = nc0;
        if (n < Nstore) {
            float blo = bias[n];
            float v0 = c00[r] + blo;
            float v1 = c10[r] + blo;
            if (GELU) { v0 = gelu_erf(v0); v1 = gelu_erf(v1); }
            D[(size_t)rlo * ldd + n] = v0;
            D[(size_t)rhi * ldd + n] = v1;
        }
        n = nc0 + 16;
        if (n < Nstore) {
            float bhi = bias[n];
            float v0 = c01[r] + bhi;
            float v1 = c11[r] + bhi;
            if (GELU) { v0 = gelu_erf(v0); v1 = gelu_erf(v1); }
            D[(size_t)rlo * ldd + n] = v0;
            D[(size_t)rhi * ldd + n] = v1;
        }
    }
}

// zero-pad W2 (512x200) -> W2p (512x256) so GEMM2 tiles cleanly by 128
__global__ __launch_bounds__(256)
void pad_w2(const float* __restrict__ W2, float* __restrict__ W2p)
{
    int idx = blockIdx.x * 256 + threadIdx.x;   // 512*256 total
    int k = idx >> 8;
    int n = idx & 255;
    W2p[idx] = (n < 200) ? W2[k * 200 + n] : 0.0f;
}

// =====================================================================
// Fused tail: overlap-add (windowed) -> conv1(+gelu) -> conv2(+gelu)
//             -> conv3 -> +residual -> pointwise conv -> out
// One block per (batch sample b, 200-wide L tile); everything in LDS.
// Halo: 21 (3 stacked K=15 convs). Zero padding at global edges is applied
// per conv layer (r1/r2 forced to 0 outside [0,2000) to match torch pad).
// =====================================================================
__global__ __launch_bounds__(256)
void fused_tail(const float* __restrict__ patches,  // ((b*11+p)*8+c) x 200
                const float* __restrict__ rw1, const float* __restrict__ rb1,
                const float* __restrict__ rw2, const float* __restrict__ rb2,
                const float* __restrict__ rw3, const float* __restrict__ rb3,
                const float* __restrict__ pw,  const float* __restrict__ pb,
                float* __restrict__ out)
{
    __shared__ float recs[8][244];   // rec on [t0-21, t0+221)
    __shared__ float r1s[32][228];   // r1  on [t0-14, t0+214)
    __shared__ float r2s[16][214];   // r2  on [t0-7,  t0+207)
    __shared__ float ss[8][200];     // rec + r3 on [t0, t0+200)

    const int tid = threadIdx.x;
    const int b   = blockIdx.x / 10;
    const int t0  = (blockIdx.x % 10) * 200;

    // ---- step 1: windowed overlap-add / wsum ----
    for (int idx = tid; idx < 8 * 242; idx += 256) {
        int c  = idx / 242;
        int xx = idx - c * 242;
        int x  = t0 - 21 + xx;
        float rv = 0.0f;
        if (x >= 0 && x < 2000) {
            int imin = (x <= 199) ? 0 : ((x - 199) + 179) / 180;
            int imax = x / 180; if (imax > 10) imax = 10;
            float acc = 0.0f, wsum = 0.0f;
            for (int i = imin; i <= imax; ++i) {
                int j = x - 180 * i;                 // 0..199
                float wv;
                if (j < 50)        wv = (float)j        * (1.0f / 49.0f);
                else if (j >= 150) wv = (float)(199 - j) * (1.0f / 49.0f);
                else               wv = 1.0f;
                acc  += patches[((size_t)(b * 11 + i) * 8 + c) * 200 + j] * wv;
                wsum += wv;
            }
            rv = acc / fmaxf(wsum, 1e-8f);
        }
        recs[c][xx] = rv;
    }
    __syncthreads();

    // ---- step 2: conv1 (8->32, K=15, pad7) + gelu ----
    for (int idx = tid; idx < 32 * 228; idx += 256) {
        int oc = idx / 228;
        int yy = idx - oc * 228;
        int y  = t0 - 14 + yy;
        float v = 0.0f;
        if (y >= 0 && y < 2000) {
            float acc = rb1[oc];
            const float* wrow = rw1 + oc * 8 * 15;
            #pragma unroll
            for (int ic = 0; ic < 8; ++ic)
                #pragma unroll
                for (int k = 0; k < 15; ++k)
                    acc = fmaf(wrow[ic * 15 + k], recs[ic][yy + k], acc);
            v = gelu_erf(acc);
        }
        r1s[oc][yy] = v;
    }
    __syncthreads();

    // ---- step 3: conv2 (32->16, K=15, pad7) + gelu ----
    for (int idx = tid; idx < 16 * 214; idx += 256) {
        int oc = idx / 214;
        int zz = idx - oc * 214;
        int z  = t0 - 7 + zz;
        float v = 0.0f;
        if (z >= 0 && z < 2000) {
            float acc = rb2[oc];
            const float* wrow = rw2 + oc * 32 * 15;
            for (int ic = 0; ic < 32; ++ic)
                #pragma unroll
                for (int k = 0; k < 15; ++k)
                    acc = fmaf(wrow[ic * 15 + k], r1s[ic][zz + k], acc);
            v = gelu_erf(acc);
        }
        r2s[oc][zz] = v;
    }
    __syncthreads();

    // ---- step 4: conv3 (16->8, K=15, pad7) + residual ----
    for (int idx = tid; idx < 8 * 200; idx += 256) {
        int c  = idx / 200;
        int xx = idx - c * 200;
        float acc = rb3[c];
        const float* wrow = rw3 + c * 16 * 15;
        for (int ic = 0; ic < 16; ++ic)
            #pragma unroll
            for (int k = 0; k < 15; ++k)
                acc = fmaf(wrow[ic * 15 + k], r2s[ic][xx + k], acc);
        ss[c][xx] = acc + recs[c][xx + 21];
    }
    __syncthreads();

    // ---- step 5: pointwise conv (8->8, K=1) ----
    for (int idx = tid; idx < 8 * 200; idx += 256) {
        int c  = idx / 200;
        int xx = idx - c * 200;
        float acc = pb[c];
        #pragma unroll
        for (int ic = 0; ic < 8; ++ic)
            acc = fmaf(pw[c * 8 + ic], ss[ic][xx], acc);
        out[((size_t)b * 8 + c) * 2000 + t0 + xx] = acc;
    }
}

extern "C" void kernel_launch(void* const* d_in, const int* in_sizes, int n_in,
                              void* d_out, int out_size, void* d_ws, size_t ws_size,
                              hipStream_t stream)
{
    (void)in_sizes; (void)n_in; (void)out_size; (void)ws_size;

    const float* E   = (const float*)d_in[0];   // (1024,11,8,512)
    const float* W1  = (const float*)d_in[1];   // (512,512)
    const float* b1  = (const float*)d_in[2];   // (512)
    const float* W2  = (const float*)d_in[3];   // (512,200)
    const float* b2  = (const float*)d_in[4];   // (200)
    const float* rw1 = (const float*)d_in[5];
    const float* rb1 = (const float*)d_in[6];
    const float* rw2 = (const float*)d_in[7];
    const float* rb2 = (const float*)d_in[8];
    const float* rw3 = (const float*)d_in[9];
    const float* rb3 = (const float*)d_in[10];
    const float* pw  = (const float*)d_in[11];
    const float* pb  = (const float*)d_in[12];
    float* out = (float*)d_out;

    // workspace layout (floats): h | W2pad | patches
    float* ws      = (float*)d_ws;
    float* h       = ws;                                    // 90112*512
    float* W2p     = ws + (size_t)90112 * 512;              // 512*256
    float* patches = W2p + (size_t)512 * 256;               // 90112*200

    pad_w2<<<512, 256, 0, stream>>>(W2, W2p);

    // GEMM1: h = GELU(E @ W1 + b1)   (90112 x 512 x 512)
    dim3 g1(90112 / 64, 512 / 128);
    gemm_f32_wmma<true><<<g1, 256, 0, stream>>>(E, W1, b1, h,
                                                512, 512, 512, 512);

    // GEMM2: patches = h @ W2 + b2   (90112 x 512 x 200, N padded to 256)
    dim3 g2(90112 / 64, 256 / 128);
    gemm_f32_wmma<false><<<g2, 256, 0, stream>>>(h, W2p, b2, patches,
                                                 512, 256, 200, 200);

    // fused overlap-add + conv refinement + residual + pointwise
    fused_tail<<<1024 * 10, 256, 0, stream>>>(patches, rw1, rb1, rw2, rb2,
                                              rw3, rb3, pw, pb, out);
}